// MoEExperts_257698038434
// MI455X (gfx1250) — compile-verified
//
#include <hip/hip_runtime.h>
#include <hip/hip_bf16.h>

// MoE experts, fp32, WMMA f32_16x16x4 on gfx1250 (wave32), with CDNA5
// async global->LDS staging (ASYNCcnt) and double-buffered LDS tiles.
//
//   T=4096 tokens, E=8 experts, H=1024, Q=1024
//   hidden'[t, e*Q+q] = rw[t,e] * silu(x@Wg[e])[t,q] * (x@Wu[e])[t,q]   (kernel 1, ws)
//   out[t,h]          = sum_{e,q} hidden'[t, e*Q+q] * down[e,q,h]       (kernel 2)

#define T_DIM 4096
#define E_DIM 8
#define H_DIM 1024
#define Q_DIM 1024

#define BK   16      // K-slab staged in LDS (4 WMMA k-steps of 4)
#define LDA  20      // padded A row stride (floats): 80B rows, 16B aligned, conflict-free
#define LDB  132     // padded B row stride (floats): 528B rows, 16B aligned

typedef float v2f __attribute__((ext_vector_type(2)));
typedef float v4f __attribute__((ext_vector_type(4)));
typedef float v8f __attribute__((ext_vector_type(8)));
typedef int   v4i __attribute__((ext_vector_type(4)));

// ---- CDNA5 async global->LDS (GLOBAL_LOAD_ASYNC_TO_LDS_B128, ASYNCcnt) ----
#if __has_builtin(__builtin_amdgcn_global_load_async_to_lds_b128) && \
    __has_builtin(__builtin_amdgcn_s_wait_asynccnt)
#define USE_ASYNC_LDS 1
typedef __attribute__((address_space(1))) v4i gbl_v4i_t;   // prints as "__device__"
typedef __attribute__((address_space(3))) v4i lds_v4i_t;   // prints as "__shared__"
#else
#define USE_ASYNC_LDS 0
#endif

__device__ __forceinline__ void copy16_to_lds(const float* g, float* l) {
#if USE_ASYNC_LDS
    __builtin_amdgcn_global_load_async_to_lds_b128((gbl_v4i_t*)g, (lds_v4i_t*)l,
                                                   /*imm offset*/0, /*cpol*/0);
#else
    *(v4f*)l = *(const v4f*)g;
#endif
}

__device__ __forceinline__ void wait_stage() {
#if USE_ASYNC_LDS
    __builtin_amdgcn_s_wait_asynccnt(0);   // own async writes landed in LDS
#endif
    __syncthreads();                       // publish to all waves in the WG
}

__device__ __forceinline__ v8f wmma_f32(v2f a, v2f b, v8f c) {
    // D = A(16x4,f32) * B(4x16,f32) + C(16x16,f32)
    return __builtin_amdgcn_wmma_f32_16x16x4_f32(false, a, false, b,
                                                 (short)0, c, false, false);
}

// ---------------------------------------------------------------------------
// Kernel 1: per expert, block computes a 128(token) x 64(q) tile of hidden'.
// Staged B tile is 128 wide: cols [0,64) = gate cols [qb,qb+64),
// cols [64,128) = up cols [Q+qb,Q+qb+64) -> gate/up accumulators for the same
// (t,q) land in the same lanes, so silu*up is a pure elementwise VALU combine.
// grid = (32*16, 8), block = 256 (8 waves: 4 along M x 2 along q)
// ---------------------------------------------------------------------------
__global__ __launch_bounds__(256)
void moe_gateup_silu_kernel(const float* __restrict__ x,      // (T,H)
                            const float* __restrict__ rw,     // (T,E)
                            const float* __restrict__ gup,    // (E,H,2Q)
                            float* __restrict__ hidden)       // (T, E*Q)
{
    __shared__ float As[2][128 * LDA];
    __shared__ float Bs[2][BK * LDB];

    const int e  = blockIdx.y;
    const int bm = blockIdx.x >> 4;     // 0..31 token tiles
    const int bq = blockIdx.x & 15;     // 0..15 q tiles
    const int t0 = bm * 128;
    const int qb = bq * 64;

    const int tid   = threadIdx.x;
    const int lane  = tid & 31;
    const int wave  = tid >> 5;
    const int waveM = wave & 3;         // 32 token rows per wave
    const int waveN = wave >> 2;        // 32 q cols per wave
    const int lhalf = lane >> 4;        // half-wave select (K pair / M+8)
    const int l16   = lane & 15;

    const float* gupE = gup + (size_t)e * H_DIM * (2 * Q_DIM);

    // k-invariant staging addresses: A two b128/thread, B two b128/thread
    const int ra = tid >> 2, ca = (tid & 3) * 4;     // 64 rows x 4 cols(x4f)
    const int rb = tid >> 4, cb = (tid & 15) * 4;    // 16 rows x 16 cols(x4f)
    const float* gA0 = x + (size_t)(t0 + ra)      * H_DIM + ca;
    const float* gA1 = x + (size_t)(t0 + ra + 64) * H_DIM + ca;
    const float* gB  = gupE + (size_t)rb * (2 * Q_DIM) + qb + cb;

    v8f acc[2][4];                      // [mtile][j]: j=0,1 gate; j=2,3 up
    #pragma unroll
    for (int i = 0; i < 2; ++i)
        #pragma unroll
        for (int j = 0; j < 4; ++j)
            #pragma unroll
            for (int k = 0; k < 8; ++k) acc[i][j][k] = 0.0f;

    auto stage = [&](int buf, int kt) {
        const int k0 = kt * BK;
        copy16_to_lds(gA0 + k0, &As[buf][ra * LDA + ca]);
        copy16_to_lds(gA1 + k0, &As[buf][(ra + 64) * LDA + ca]);
        const float* gBr = gB + (size_t)k0 * (2 * Q_DIM);
        copy16_to_lds(gBr,         &Bs[buf][rb * LDB + cb]);
        copy16_to_lds(gBr + Q_DIM, &Bs[buf][rb * LDB + 64 + cb]);
    };

    stage(0, 0);
    wait_stage();

    const int NK = H_DIM / BK;
    for (int kt = 0; kt < NK; ++kt) {
        const int buf = kt & 1;
        if (kt + 1 < NK) stage(buf ^ 1, kt + 1);   // overlap with compute below

        const float* Asb = As[buf];
        const float* Bsb = Bs[buf];
        #pragma unroll
        for (int ks = 0; ks < 4; ++ks) {
            const int kk = ks * 4 + 2 * lhalf;  // lanes 0-15: K{0,1}; 16-31: K{2,3}
            v2f afrag[2];
            #pragma unroll
            for (int mt = 0; mt < 2; ++mt) {
                const int row = waveM * 32 + mt * 16 + l16;
                afrag[mt] = *(const v2f*)&Asb[row * LDA + kk];
            }
            #pragma unroll
            for (int j = 0; j < 4; ++j) {
                const int col = (j >> 1) * 64 + waveN * 32 + (j & 1) * 16 + l16;
                v2f bfrag;
                bfrag.x = Bsb[(kk)     * LDB + col];
                bfrag.y = Bsb[(kk + 1) * LDB + col];
                #pragma unroll
                for (int mt = 0; mt < 2; ++mt)
                    acc[mt][j] = wmma_f32(afrag[mt], bfrag, acc[mt][j]);
            }
        }
        wait_stage();
    }

    // Epilogue: hidden' = rw * silu(gate) * up.  C/D layout: VGPR v holds
    // M = v + 8*lhalf, N = l16.
    #pragma unroll
    for (int mt = 0; mt < 2; ++mt) {
        #pragma unroll
        for (int nt = 0; nt < 2; ++nt) {
            const v8f g = acc[mt][nt];
            const v8f u = acc[mt][nt + 2];
            const int q = qb + waveN * 32 + nt * 16 + l16;
            #pragma unroll
            for (int v = 0; v < 8; ++v) {
                const int t = t0 + waveM * 32 + mt * 16 + v + 8 * lhalf;
                const float gv = g[v];
                const float hv = (gv / (1.0f + __expf(-gv))) * u[v];
                const float w  = rw[t * E_DIM + e];
                hidden[(size_t)t * (E_DIM * Q_DIM) + e * Q_DIM + q] = hv * w;
            }
        }
    }
}

// ---------------------------------------------------------------------------
// Kernel 2: out(4096x1024) = hidden'(4096x8192) @ down_flat(8192x1024).
// grid = (8, 32), block = 256 (8 waves: 4 along M x 2 along N; wave = 32x64)
// ---------------------------------------------------------------------------
__global__ __launch_bounds__(256)
void moe_down_kernel(const float* __restrict__ hidden,   // (T, E*Q)
                     const float* __restrict__ down,     // (E*Q, H) row-major
                     float* __restrict__ out)            // (T, H)
{
    __shared__ float As[2][128 * LDA];
    __shared__ float Bs[2][BK * LDB];

    const int n0 = blockIdx.x * 128;
    const int t0 = blockIdx.y * 128;
    const int KTOT = E_DIM * Q_DIM;     // 8192

    const int tid   = threadIdx.x;
    const int lane  = tid & 31;
    const int wave  = tid >> 5;
    const int waveM = wave & 3;         // 32 rows
    const int waveN = wave >> 2;        // 64 cols (4 subtiles)
    const int lhalf = lane >> 4;
    const int l16   = lane & 15;

    const int ra = tid >> 2, ca = (tid & 3) * 4;
    const int rb = tid >> 4, cb = (tid & 15) * 4;
    const float* gA0 = hidden + (size_t)(t0 + ra)      * KTOT + ca;
    const float* gA1 = hidden + (size_t)(t0 + ra + 64) * KTOT + ca;
    const float* gB  = down + (size_t)rb * H_DIM + n0;

    v8f acc[2][4];
    #pragma unroll
    for (int i = 0; i < 2; ++i)
        #pragma unroll
        for (int j = 0; j < 4; ++j)
            #pragma unroll
            for (int k = 0; k < 8; ++k) acc[i][j][k] = 0.0f;

    auto stage = [&](int buf, int kt) {
        const int k0 = kt * BK;
        copy16_to_lds(gA0 + k0, &As[buf][ra * LDA + ca]);
        copy16_to_lds(gA1 + k0, &As[buf][(ra + 64) * LDA + ca]);
        const float* gBr = gB + (size_t)k0 * H_DIM;
        copy16_to_lds(gBr + cb,      &Bs[buf][rb * LDB + cb]);
        copy16_to_lds(gBr + 64 + cb, &Bs[buf][rb * LDB + 64 + cb]);
    };

    stage(0, 0);
    wait_stage();

    const int NK = KTOT / BK;           // 512
    for (int kt = 0; kt < NK; ++kt) {
        const int buf = kt & 1;
        if (kt + 1 < NK) stage(buf ^ 1, kt + 1);

        const float* Asb = As[buf];
        const float* Bsb = Bs[buf];
        #pragma unroll
        for (int ks = 0; ks < 4; ++ks) {
            const int kk = ks * 4 + 2 * lhalf;
            v2f afrag[2];
            #pragma unroll
            for (int mt = 0; mt < 2; ++mt) {
                const int row = waveM * 32 + mt * 16 + l16;
                afrag[mt] = *(const v2f*)&Asb[row * LDA + kk];
            }
            #pragma unroll
            for (int j = 0; j < 4; ++j) {
                const int col = waveN * 64 + j * 16 + l16;
                v2f bfrag;
                bfrag.x = Bsb[(kk)     * LDB + col];
                bfrag.y = Bsb[(kk + 1) * LDB + col];
                #pragma unroll
                for (int mt = 0; mt < 2; ++mt)
                    acc[mt][j] = wmma_f32(afrag[mt], bfrag, acc[mt][j]);
            }
        }
        wait_stage();
    }

    #pragma unroll
    for (int mt = 0; mt < 2; ++mt) {
        #pragma unroll
        for (int j = 0; j < 4; ++j) {
            const int n = n0 + waveN * 64 + j * 16 + l16;
            #pragma unroll
            for (int v = 0; v < 8; ++v) {
                const int t = t0 + waveM * 32 + mt * 16 + v + 8 * lhalf;
                out[(size_t)t * H_DIM + n] = acc[mt][j][v];
            }
        }
    }
}

extern "C" void kernel_launch(void* const* d_in, const int* in_sizes, int n_in,
                              void* d_out, int out_size, void* d_ws, size_t ws_size,
                              hipStream_t stream) {
    const float* x   = (const float*)d_in[0];   // (T,H)
    const float* rw  = (const float*)d_in[1];   // (T,E)
    const float* gup = (const float*)d_in[2];   // (E,H,2Q)
    const float* dwn = (const float*)d_in[3];   // (E,Q,H) == (E*Q, H) row-major
    float* out    = (float*)d_out;              // (T,H)
    float* hidden = (float*)d_ws;               // needs T*E*Q*4 = 128 MiB

    dim3 blk(256);
    dim3 g1(32 * 16, E_DIM);                    // token tiles x q tiles, per expert
    hipLaunchKernelGGL(moe_gateup_silu_kernel, g1, blk, 0, stream, x, rw, gup, hidden);

    dim3 g2(H_DIM / 128, T_DIM / 128);
    hipLaunchKernelGGL(moe_down_kernel, g2, blk, 0, stream, hidden, dwn, out);
}